// ClustGeoNodeEncoder_72060961292776
// MI455X (gfx1250) — compile-verified
//
#include <hip/hip_runtime.h>
#include <math.h>

// CDNA5 / gfx1250, wave32.
typedef __attribute__((ext_vector_type(2))) float v2f;
typedef __attribute__((ext_vector_type(8))) float v8f;

#define N_CLUSTS   32768
#define CLUST_SIZE 128
#define WAVES_PER_BLOCK 8
#define BLOCK_THREADS   (WAVES_PER_BLOCK * 32)
#define COMP_STRIDE 132            // 128 + 4 dword pad: de-alias LDS banks
#define WAVE_LDS    (4 * COMP_STRIDE)

__global__ __launch_bounds__(BLOCK_THREADS)
void clust_geo_kernel(const float* __restrict__ data,   // (N_VOX, 6) f32
                      const int*   __restrict__ clusts, // (N_CLUSTS, 128) i32
                      float*       __restrict__ out)    // (N_CLUSTS, 16) f32
{
    // Per-wave staging: component-major [4][COMP_STRIDE] floats
    // (planes x,y,z,ones) so each lane's two K-slots are one b64 apart.
    __shared__ __align__(16) float lds[WAVES_PER_BLOCK * WAVE_LDS];

    const int tid   = threadIdx.x;
    const int wave  = tid >> 5;
    const int lane  = tid & 31;
    const int cidx  = lane & 15;   // M/N index inside the 16x16 tile
    const int g     = lane >> 4;   // lane-half -> high bit of K
    const int clust = blockIdx.x * WAVES_PER_BLOCK + wave;

    float* ldsW = lds + wave * WAVE_LDS;

    // ---- Pass 1: gather 4 voxels per lane, keep in regs, stage to LDS ----
    float px[4], py[4], pz[4];
    const int base = clust * CLUST_SIZE + lane * 4;
    #pragma unroll
    for (int j = 0; j < 4; ++j) {
        const long long idx = (long long)clusts[base + j];
        const float* row = data + (size_t)idx * 6;
        px[j] = row[1];
        py[j] = row[2];
        pz[j] = row[3];
    }
    const int s4 = lane * 4;       // this lane's 4 contiguous samples
    *(float4*)(ldsW + 0 * COMP_STRIDE + s4) = make_float4(px[0], px[1], px[2], px[3]);
    *(float4*)(ldsW + 1 * COMP_STRIDE + s4) = make_float4(py[0], py[1], py[2], py[3]);
    *(float4*)(ldsW + 2 * COMP_STRIDE + s4) = make_float4(pz[0], pz[1], pz[2], pz[3]);
    *(float4*)(ldsW + 3 * COMP_STRIDE + s4) = make_float4(1.0f, 1.0f, 1.0f, 1.0f);
    __syncthreads();

    // ---- Homogeneous moment matrix S~ = sum_s [x y z 1]^T [x y z 1] ----
    // 32 V_WMMA_F32_16X16X4_F32 on 4 independent accumulator chains.
    // A and B get the same per-lane tile (A layout: lane=M, K=2*lanehalf+vgpr;
    // B uses the same K packing, so B = A^T and D accumulates X_t^T X_t).
    // Lanes cidx>=4 read the ones-plane; that only pollutes D rows/cols >= 4.
    const int    mc      = (cidx < 3) ? cidx : 3;
    const float* lanePtr = ldsW + mc * COMP_STRIDE + 2 * g;  // + 4*t per step

    // Load ALL K-tiles into distinct registers first so the WMMA chain never
    // stalls on a fresh ds_load (waits become incremental, not dscnt==0).
    v2f a[32];
    #pragma unroll
    for (int t = 0; t < 32; ++t)
        a[t] = *(const v2f*)(lanePtr + 4 * t);

    v8f acc0 = {}, acc1 = {}, acc2 = {}, acc3 = {};
    #pragma unroll
    for (int t = 0; t < 32; t += 4) {
        acc0 = __builtin_amdgcn_wmma_f32_16x16x4_f32(false, a[t+0], false, a[t+0], (short)0, acc0, false, false);
        acc1 = __builtin_amdgcn_wmma_f32_16x16x4_f32(false, a[t+1], false, a[t+1], (short)0, acc1, false, false);
        acc2 = __builtin_amdgcn_wmma_f32_16x16x4_f32(false, a[t+2], false, a[t+2], (short)0, acc2, false, false);
        acc3 = __builtin_amdgcn_wmma_f32_16x16x4_f32(false, a[t+3], false, a[t+3], (short)0, acc3, false, false);
    }
    const v8f acc = (acc0 + acc1) + (acc2 + acc3);

    // D layout: vgpr i, lane j (j<16) holds D[i][j]. Broadcast the 4x4 block.
    const float S00 = __shfl(acc[0], 0, 32);
    const float S01 = __shfl(acc[0], 1, 32);
    const float S02 = __shfl(acc[0], 2, 32);
    const float S03 = __shfl(acc[0], 3, 32);
    const float S11 = __shfl(acc[1], 1, 32);
    const float S12 = __shfl(acc[1], 2, 32);
    const float S13 = __shfl(acc[1], 3, 32);
    const float S22 = __shfl(acc[2], 2, 32);
    const float S23 = __shfl(acc[2], 3, 32);
    const float S33 = __shfl(acc[3], 3, 32);   // == 128.0f (count)

    const float n     = S33;
    const float inv_n = 1.0f / n;
    const float cx = S03 * inv_n, cy = S13 * inv_n, cz = S23 * inv_n;

    // Covariance (un-normalized, like the reference): A = S - n * c c^T
    const float A00 = S00 - n * cx * cx;
    const float A01 = S01 - n * cx * cy;
    const float A02 = S02 - n * cx * cz;
    const float A11 = S11 - n * cy * cy;
    const float A12 = S12 - n * cy * cz;
    const float A22 = S22 - n * cz * cz;

    // ---- Closed-form symmetric 3x3 eigenvalues (trig method) ----
    const float q   = (A00 + A11 + A22) * (1.0f / 3.0f);
    const float p1  = A01 * A01 + A02 * A02 + A12 * A12;
    const float a0q = A00 - q, a1q = A11 - q, a2q = A22 - q;
    const float p2  = a0q * a0q + a1q * a1q + a2q * a2q + 2.0f * p1;
    const float p   = sqrtf(fmaxf(p2, 0.0f) * (1.0f / 6.0f));
    const float invp = 1.0f / fmaxf(p, 1e-20f);
    const float b00 = a0q * invp, b11 = a1q * invp, b22 = a2q * invp;
    const float b01 = A01 * invp, b02 = A02 * invp, b12 = A12 * invp;
    const float detB = b00 * (b11 * b22 - b12 * b12)
                     - b01 * (b01 * b22 - b12 * b02)
                     + b02 * (b01 * b12 - b11 * b02);
    const float r   = fminf(fmaxf(0.5f * detB, -1.0f), 1.0f);
    const float phi = acosf(r) * (1.0f / 3.0f);
    const float TWO_PI_3 = 2.09439510239319549f;
    const float w2 = q + 2.0f * p * cosf(phi);             // largest
    const float w0 = q + 2.0f * p * cosf(phi + TWO_PI_3);  // smallest
    const float w1 = 3.0f * q - w2 - w0;                   // middle
    const float w2s   = (fabsf(w2) > 1e-30f) ? w2 : 1e-30f;
    const float dirwt = 1.0f - w1 / w2s;
    const float invw2 = 1.0f / w2s;

    // ---- Eigenvector of w2: largest-norm cross of rows of (A - w2 I) ----
    const float m00 = A00 - w2, m11 = A11 - w2, m22 = A22 - w2;
    const float v01x = A01 * A12 - A02 * m11;   // r0 x r1
    const float v01y = A02 * A01 - m00 * A12;
    const float v01z = m00 * m11 - A01 * A01;
    const float v02x = A01 * m22 - A02 * A12;   // r0 x r2
    const float v02y = A02 * A02 - m00 * m22;
    const float v02z = m00 * A12 - A01 * A02;
    const float v12x = m11 * m22 - A12 * A12;   // r1 x r2
    const float v12y = A12 * A02 - A01 * m22;
    const float v12z = A01 * A12 - m11 * A02;
    const float n01 = v01x * v01x + v01y * v01y + v01z * v01z;
    const float n02 = v02x * v02x + v02y * v02y + v02z * v02z;
    const float n12 = v12x * v12x + v12y * v12y + v12z * v12z;
    float vx = v01x, vy = v01y, vz = v01z, nb = n01;
    if (n02 > nb) { vx = v02x; vy = v02y; vz = v02z; nb = n02; }
    if (n12 > nb) { vx = v12x; vy = v12y; vz = v12z; nb = n12; }
    const float invl = 1.0f / sqrtf(fmaxf(nb, 1e-30f));
    vx *= invl; vy *= invl; vz *= invl;

    // ---- Pass 2 (register-resident): sc = sum x0 * ||xc - x0 v|| ----
    float scp = 0.0f;
    #pragma unroll
    for (int j = 0; j < 4; ++j) {
        const float xcx = px[j] - cx, xcy = py[j] - cy, xcz = pz[j] - cz;
        const float x0  = xcx * vx + xcy * vy + xcz * vz;
        const float xpx = xcx - x0 * vx;
        const float xpy = xcy - x0 * vy;
        const float xpz = xcz - x0 * vz;
        const float np0 = sqrtf(xpx * xpx + xpy * xpy + xpz * xpz);
        scp = fmaf(x0, np0, scp);
    }
    #pragma unroll
    for (int off = 16; off > 0; off >>= 1)
        scp += __shfl_xor(scp, off, 32);
    const float sgn = (scp < 0.0f) ? -dirwt : dirwt;

    // ---- Coalesced 16-float output per cluster (lanes 0..15) ----
    float ov;
    const int k = lane;
    if      (k == 0)  ov = cx;
    else if (k == 1)  ov = cy;
    else if (k == 2)  ov = cz;
    else if (k == 3)  ov = A00 * invw2;
    else if (k == 4)  ov = A01 * invw2;
    else if (k == 5)  ov = A02 * invw2;
    else if (k == 6)  ov = A01 * invw2;
    else if (k == 7)  ov = A11 * invw2;
    else if (k == 8)  ov = A12 * invw2;
    else if (k == 9)  ov = A02 * invw2;
    else if (k == 10) ov = A12 * invw2;
    else if (k == 11) ov = A22 * invw2;
    else if (k == 12) ov = vx * sgn;
    else if (k == 13) ov = vy * sgn;
    else if (k == 14) ov = vz * sgn;
    else              ov = (float)CLUST_SIZE;
    if (lane < 16)
        out[(size_t)clust * 16 + lane] = ov;
}

extern "C" void kernel_launch(void* const* d_in, const int* in_sizes, int n_in,
                              void* d_out, int out_size, void* d_ws, size_t ws_size,
                              hipStream_t stream) {
    (void)in_sizes; (void)n_in; (void)out_size; (void)d_ws; (void)ws_size;
    const float* data   = (const float*)d_in[0];
    const int*   clusts = (const int*)d_in[1];
    float*       out    = (float*)d_out;
    dim3 grid(N_CLUSTS / WAVES_PER_BLOCK);
    clust_geo_kernel<<<grid, BLOCK_THREADS, 0, stream>>>(data, clusts, out);
}